// MPSAGEConv_54494545052116
// MI455X (gfx1250) — compile-verified
//
#include <hip/hip_runtime.h>

// ---- static problem geometry (matches reference) ----
#define WORLD   4
#define S_SEEDS 16000
#define FANOUT  16
#define F_FRONT (S_SEEDS * (FANOUT + 1))   // 272000 frontier rows / partition
#define E_EDGES (S_SEEDS * FANOUT)         // 256000 edges / partition
#define D       128
#define N_ROWS  (WORLD * S_SEEDS)          // 64000 output rows
#define TOT_E   (WORLD * E_EDGES)          // 1024000 edges total

typedef __attribute__((ext_vector_type(2))) float v2f;
typedef __attribute__((ext_vector_type(4))) float v4f;
typedef __attribute__((ext_vector_type(8))) float v8f;

// ------------------------------------------------------------------
// Kernel 1: zero the accumulator (d_out doubles as h_neigh buffer)
// ------------------------------------------------------------------
__global__ void sage_zero(float* __restrict__ out) {
    size_t i = (size_t)blockIdx.x * blockDim.x + threadIdx.x; // over v4f elems
    v4f z = {0.f, 0.f, 0.f, 0.f};
    ((v4f*)out)[i] = z;
}

// ------------------------------------------------------------------
// Kernel 2: edge scatter-add  out[dst] += feat[src]
// one wave per edge: 32 lanes x float4 = 128 floats
// ------------------------------------------------------------------
__global__ __launch_bounds__(256) void sage_scatter(
    const float* __restrict__ feat,
    const int*   __restrict__ coo_row,
    const int*   __restrict__ coo_col,
    float*       __restrict__ out) {
    const int wave = threadIdx.x >> 5;
    const int lane = threadIdx.x & 31;
    const int e    = blockIdx.x * 8 + wave;

    const int part = e / E_EDGES;
    const int src  = part * F_FRONT + coo_col[e];
    const int dst  = part * S_SEEDS + coo_row[e];

    // streaming read of the source feature row (low reuse -> non-temporal)
    const v4f* sp = (const v4f*)(feat + (size_t)src * D);
    v4f v = __builtin_nontemporal_load(&sp[lane]);

    float* dp = out + (size_t)dst * D + lane * 4;
    atomicAdd(dp + 0, v.x);
    atomicAdd(dp + 1, v.y);
    atomicAdd(dp + 2, v.z);
    atomicAdd(dp + 3, v.w);
}

// ------------------------------------------------------------------
// Kernel 3: fused  out = feat_dst @ Ws^T + b + (out/FANOUT) @ Wn^T
// 8 waves/block, 16 rows/wave, f32 WMMA 16x16x4, weights in swizzled LDS.
// ------------------------------------------------------------------
__global__ __launch_bounds__(256) void sage_gemm(
    const float* __restrict__ feat,
    const float* __restrict__ wn,    // fc_neigh_w [128,128] row-major [n][k]
    const float* __restrict__ wsw,   // fc_self_w  [128,128]
    const float* __restrict__ bias,  // fc_self_b  [128]
    float*       __restrict__ out) {

    // LDS holds one weight matrix, transposed (k-major) with an XOR-32 column
    // swizzle so B-fragment reads (lanes 0-15: row k, lanes 16-31: row k+2)
    // land on disjoint bank halves.  element (k,n) -> wlds[k*128 + (n ^ ((k&2)<<4))]
    __shared__ float wlds[128 * 128]; // 64 KB

    const int lane  = threadIdx.x & 31;
    const int wave  = threadIdx.x >> 5;
    const int nlane = lane & 15;   // A: row-in-tile / B,C,D: col-in-tile
    const int khalf = lane >> 4;   // 0: K0/K1 (M0..7 for C), 1: K2/K3 (M8..15)
    const int kb    = khalf * 2;
    const int m0    = blockIdx.x * 128 + wave * 16;

    // A row pointers (per lane): self rows come from feat_dst mapping
    const int   rGlob   = m0 + nlane;                 // [0, 64000)
    const int   part    = rGlob / S_SEEDS;
    const float* selfRow  = feat + ((size_t)part * F_FRONT + (rGlob - part * S_SEEDS)) * D;
    const float* neighRow = out  +  (size_t)rGlob * D;

    v8f acc[8];
#pragma unroll
    for (int t = 0; t < 8; ++t) acc[t] = (v8f){};

    // ================= pass 0: self term =================
    for (int i = threadIdx.x; i < 128 * 128; i += 256) {
        int k = i >> 7, c = i & 127;
        int n = c ^ ((k & 2) << 4);      // involution: recover source column
        wlds[i] = wsw[n * 128 + k];
    }
    __syncthreads();

#pragma unroll 2
    for (int k0 = 0; k0 < 128; k0 += 4) {
        const int kp = k0 + kb;                       // this lane's K pair base
        v2f a = *(const v2f*)(selfRow + kp);          // A[m][kp], A[m][kp+1]
#pragma unroll
        for (int n0 = 0; n0 < 8; ++n0) {
            const int n = n0 * 16 + nlane;
            v2f b;
            b.x = wlds[ kp      * 128 + (n ^ (( kp      & 2) << 4))];
            b.y = wlds[(kp + 1) * 128 + (n ^ (((kp + 1) & 2) << 4))];
            acc[n0] = __builtin_amdgcn_wmma_f32_16x16x4_f32(
                false, a, false, b, (short)0, acc[n0], false, false);
        }
    }
    __syncthreads();

    // ================= pass 1: neighbor term (1/FANOUT folded into weights) ==
    for (int i = threadIdx.x; i < 128 * 128; i += 256) {
        int k = i >> 7, c = i & 127;
        int n = c ^ ((k & 2) << 4);
        wlds[i] = wn[n * 128 + k] * (1.0f / FANOUT);
    }
    __syncthreads();

#pragma unroll 2
    for (int k0 = 0; k0 < 128; k0 += 4) {
        const int kp = k0 + kb;
        v2f a = *(const v2f*)(neighRow + kp);         // reads own rows of out
#pragma unroll
        for (int n0 = 0; n0 < 8; ++n0) {
            const int n = n0 * 16 + nlane;
            v2f b;
            b.x = wlds[ kp      * 128 + (n ^ (( kp      & 2) << 4))];
            b.y = wlds[(kp + 1) * 128 + (n ^ (((kp + 1) & 2) << 4))];
            acc[n0] = __builtin_amdgcn_wmma_f32_16x16x4_f32(
                false, a, false, b, (short)0, acc[n0], false, false);
        }
    }

    // ================= epilogue: bias + in-place store =================
    // C/D layout: VGPR j, lane L -> M = j + 8*(L>=16), N = L&15
#pragma unroll
    for (int n0 = 0; n0 < 8; ++n0) {
        const float bv = bias[n0 * 16 + nlane];
#pragma unroll
        for (int j = 0; j < 8; ++j) {
            const int mrow = m0 + j + 8 * khalf;
            out[(size_t)mrow * D + n0 * 16 + nlane] = acc[n0][j] + bv;
        }
    }
}

// ------------------------------------------------------------------
extern "C" void kernel_launch(void* const* d_in, const int* in_sizes, int n_in,
                              void* d_out, int out_size, void* d_ws, size_t ws_size,
                              hipStream_t stream) {
    (void)in_sizes; (void)n_in; (void)out_size; (void)d_ws; (void)ws_size;

    const float* feat    = (const float*)d_in[0];
    const int*   coo_row = (const int*)  d_in[1];
    const int*   coo_col = (const int*)  d_in[2];
    const float* wn      = (const float*)d_in[3];
    const float* wsw     = (const float*)d_in[4];
    const float* bias    = (const float*)d_in[5];
    float*       out     = (float*)d_out;

    // 1) zero accumulator: 64000*128 floats = 2,048,000 float4
    sage_zero<<<(N_ROWS * D / 4) / 256, 256, 0, stream>>>(out);

    // 2) scatter edges: 8 edges (waves) per 256-thread block
    sage_scatter<<<TOT_E / 8, 256, 0, stream>>>(feat, coo_row, coo_col, out);

    // 3) fused dual-GEMM + bias, 128 rows per block
    sage_gemm<<<N_ROWS / 128, 256, 0, stream>>>(feat, wn, wsw, bias, out);
}